// DualMambaBlock_69363721830802
// MI455X (gfx1250) — compile-verified
//
#include <hip/hip_runtime.h>
#include <hip/hip_bf16.h>
#include <math.h>

// DualMambaBlock for MI455X (gfx1250, wave32).
// fp32 GEMMs on the matrix pipe via V_WMMA_F32_16X16X4_F32 (keeps fp32 numerics),
// scan/conv/proj on VALU. Workspace requirement: ~181 MB (see layout below).

#define BB 4
#define LL 256
#define CC 32
#define DD 128
#define DI 256
#define DS 16
#define DCONV 4
#define DR 8
#define NTOK (BB * LL * CC)   /* 32768 tokens per mamba pass */
#define XZN (2 * DI)          /* 512 */

typedef float v2f __attribute__((ext_vector_type(2)));
typedef float v8f __attribute__((ext_vector_type(8)));

__device__ __forceinline__ float fsilu(float x) { return x / (1.0f + __expf(-x)); }

// Temporal token order is (b, c, l); x / T_out live as (b, l, c, d).
// Maps token row r -> flat element offset of that row in the (B,L,C,D) tensor.
__device__ __forceinline__ long t_row_off(int r) {
    int b   = r / (CC * LL);
    int rem = r - b * (CC * LL);
    int c   = rem / LL;
    int l   = rem - c * LL;
    return ((long)(b * LL + l) * CC + c) * DD;
}

// ---------------------------------------------------------------------------
// GEMM 1: XZ[r, 0..511] = A_row(r)[0..127] @ W(128, 512)
// grid.x = NTOK/16, block = 256 (8 waves). Wave w computes a 16x64 strip
// (4 N-tiles), reusing one A fragment per k-step across 4 WMMAs.
// mode 0: A row r at X + r*DD (contiguous). mode 1: gather from x via t_row_off.
// ---------------------------------------------------------------------------
__global__ void k_gemm_in(const float* __restrict__ X, const float* __restrict__ W,
                          float* __restrict__ XZ, int mode)
{
    const int lane = threadIdx.x & 31;
    const int wave = threadIdx.x >> 5;
    const int m0   = blockIdx.x * 16;
    const int mrow = m0 + (lane & 15);
    const int ksel = (lane >> 4) * 2;       // lanes 0-15: K0/K1, lanes 16-31: K2/K3
    const int n0   = wave * 64;
    const int col  = lane & 15;

    const long aoff = (mode == 0) ? (long)mrow * DD : t_row_off(mrow);

    v8f acc0 = {}, acc1 = {}, acc2 = {}, acc3 = {};
    for (int k0 = 0; k0 < DD; k0 += 4) {
        v2f a = *(const v2f*)(X + aoff + k0 + ksel);
        const float* wr0 = W + (long)(k0 + ksel) * XZN;
        const float* wr1 = wr0 + XZN;
        __builtin_prefetch(wr0 + 4 * XZN + n0 + col, 0, 1);
        v2f b0 = { wr0[n0 +  0 + col], wr1[n0 +  0 + col] };
        v2f b1 = { wr0[n0 + 16 + col], wr1[n0 + 16 + col] };
        v2f b2 = { wr0[n0 + 32 + col], wr1[n0 + 32 + col] };
        v2f b3 = { wr0[n0 + 48 + col], wr1[n0 + 48 + col] };
        acc0 = __builtin_amdgcn_wmma_f32_16x16x4_f32(false, a, false, b0, (short)0, acc0, false, false);
        acc1 = __builtin_amdgcn_wmma_f32_16x16x4_f32(false, a, false, b1, (short)0, acc1, false, false);
        acc2 = __builtin_amdgcn_wmma_f32_16x16x4_f32(false, a, false, b2, (short)0, acc2, false, false);
        acc3 = __builtin_amdgcn_wmma_f32_16x16x4_f32(false, a, false, b3, (short)0, acc3, false, false);
    }
    // C/D layout: VGPR i, lanes 0-15 -> M=i, lanes 16-31 -> M=8+i; N = lane&15.
    const int rbase = m0 + ((lane >> 4) ? 8 : 0);
#pragma unroll
    for (int i = 0; i < 8; ++i) {
        float* o = XZ + (long)(rbase + i) * XZN + n0 + col;
        o[ 0] = acc0[i];
        o[16] = acc1[i];
        o[32] = acc2[i];
        o[48] = acc3[i];
    }
}

// ---------------------------------------------------------------------------
// GEMM 2: OUT_row(r)[0..127] = YG[r, 0..255] @ W(256, 128)
// grid.x = NTOK/16, block = 256 (8 waves); wave w owns N-tile n0 = w*16.
// mode 0: contiguous rows at OUT + r*DD. mode 1: temporal scatter via t_row_off.
// ---------------------------------------------------------------------------
__global__ void k_gemm_out(const float* __restrict__ YG, const float* __restrict__ W,
                           float* __restrict__ OUT, int mode)
{
    const int lane = threadIdx.x & 31;
    const int wave = threadIdx.x >> 5;
    const int m0   = blockIdx.x * 16;
    const int mrow = m0 + (lane & 15);
    const int ksel = (lane >> 4) * 2;
    const int n0   = wave * 16;
    const int col  = lane & 15;

    const long aoff = (long)mrow * DI;
    v8f acc = {};
    for (int k0 = 0; k0 < DI; k0 += 4) {
        v2f a = *(const v2f*)(YG + aoff + k0 + ksel);
        __builtin_prefetch(YG + aoff + k0 + 8, 0, 1);
        v2f b = { W[(long)(k0 + ksel) * DD + n0 + col],
                  W[(long)(k0 + ksel + 1) * DD + n0 + col] };
        acc = __builtin_amdgcn_wmma_f32_16x16x4_f32(false, a, false, b, (short)0, acc, false, false);
    }
    const int rbase = m0 + ((lane >> 4) ? 8 : 0);
#pragma unroll
    for (int i = 0; i < 8; ++i) {
        int  row = rbase + i;
        long off = (mode == 1) ? t_row_off(row) : (long)row * DD;
        OUT[off + n0 + col] = acc[i];
    }
}

// ---------------------------------------------------------------------------
// Depthwise causal conv (width 4) + bias + SiLU on the u half of xz.
// grid.x = NTOK, block = DI. conv_w layout (DC, 1, DI).
// ---------------------------------------------------------------------------
__global__ void k_conv_silu(const float* __restrict__ XZ, const float* __restrict__ cw,
                            const float* __restrict__ cb, float* __restrict__ U, int seqlen)
{
    const int r   = blockIdx.x;
    const int ch  = threadIdx.x;
    const int seq = r / seqlen;
    const int t   = r - seq * seqlen;
    float acc = cb[ch];
#pragma unroll
    for (int j = 0; j < DCONV; ++j) {
        int tt = t - (DCONV - 1) + j;
        if (tt >= 0)
            acc += cw[j * DI + ch] * XZ[(long)(seq * seqlen + tt) * XZN + ch];
    }
    U[(long)r * DI + ch] = fsilu(acc);
}

// ---------------------------------------------------------------------------
// x_dbl = u @ w_xproj (256 -> 40); dt = softplus(x_dbl[:8] @ w_dt + b_dt);
// B = x_dbl[8:24], C = x_dbl[24:40] packed into BCbuf rows of 32.
// grid.x = NTOK, block = DI.
// ---------------------------------------------------------------------------
__global__ void k_xproj_dt(const float* __restrict__ U, const float* __restrict__ wx,
                           const float* __restrict__ wdt, const float* __restrict__ bdt,
                           float* __restrict__ DT, float* __restrict__ BCbuf)
{
    __shared__ float su[DI];
    __shared__ float sx[DR + 2 * DS];   // 40
    const int r   = blockIdx.x;
    const int tid = threadIdx.x;

    su[tid] = U[(long)r * DI + tid];
    __syncthreads();
    if (tid < DR + 2 * DS) {
        float a = 0.0f;
        for (int k = 0; k < DI; ++k) a += su[k] * wx[k * (DR + 2 * DS) + tid];
        sx[tid] = a;
    }
    __syncthreads();
    if (tid < 2 * DS) BCbuf[(long)r * 32 + tid] = sx[DR + tid];

    float a = bdt[tid];
#pragma unroll
    for (int j = 0; j < DR; ++j) a += sx[j] * wdt[j * DI + tid];
    DT[(long)r * DI + tid] = (a > 20.0f) ? a : __logf(1.0f + __expf(a));
}

// ---------------------------------------------------------------------------
// Selective scan. One block per sequence; thread = DI channel, 16 states in
// registers, B_t/C_t broadcast via LDS. Fuses +u*D_skip and the silu(z) gate.
// grid.x = nseq, block = DI.
// ---------------------------------------------------------------------------
__global__ void k_scan(const float* __restrict__ U, const float* __restrict__ DT,
                       const float* __restrict__ BCbuf, const float* __restrict__ XZ,
                       const float* __restrict__ alog, const float* __restrict__ dsk,
                       float* __restrict__ YG, int seqlen)
{
    __shared__ float bc[2 * DS];
    const int seq = blockIdx.x;
    const int di  = threadIdx.x;

    float A[DS], h[DS];
#pragma unroll
    for (int s = 0; s < DS; ++s) { A[s] = -__expf(alog[di * DS + s]); h[s] = 0.0f; }
    const float skip = dsk[di];

    for (int t = 0; t < seqlen; ++t) {
        const long r = (long)seq * seqlen + t;
        __syncthreads();
        if (di < 2 * DS) bc[di] = BCbuf[r * 32 + di];
        __syncthreads();
        float dt = DT[r * DI + di];
        float u  = U[r * DI + di];
        float du = dt * u;
        float y  = 0.0f;
#pragma unroll
        for (int s = 0; s < DS; ++s) {
            h[s] = h[s] * __expf(dt * A[s]) + du * bc[s];
            y += h[s] * bc[DS + s];
        }
        float z = XZ[r * XZN + DI + di];
        YG[r * DI + di] = (y + u * skip) * fsilu(z);
    }
}

// ---------------------------------------------------------------------------
// node = mean_L(x) @ g_w_node + g_b_node.  grid.x = B*C, block = DD.
// ---------------------------------------------------------------------------
__global__ void k_node(const float* __restrict__ x, const float* __restrict__ wn,
                       const float* __restrict__ bn, float* __restrict__ node)
{
    __shared__ float sm[DD];
    const int bc = blockIdx.x;
    const int b  = bc / CC, c = bc - b * CC;
    const int d  = threadIdx.x;
    float s = 0.0f;
    for (int l = 0; l < LL; ++l) s += x[(((long)b * LL + l) * CC + c) * DD + d];
    sm[d] = s * (1.0f / LL);
    __syncthreads();
    if (d < DD / 2) {
        float a = bn[d];
        for (int k = 0; k < DD; ++k) a += sm[k] * wn[k * (DD / 2) + d];
        node[(long)bc * (DD / 2) + d] = a;
    }
}

// P = sigmoid(node @ node^T / sqrt(64)).  grid.x = B, block = 256.
__global__ void k_attn(const float* __restrict__ node, float* __restrict__ P)
{
    __shared__ float sn[CC * (DD / 2)];   // 2048 floats
    const int b   = blockIdx.x;
    const int tid = threadIdx.x;
    for (int i = tid; i < CC * (DD / 2); i += 256) sn[i] = node[(long)b * CC * (DD / 2) + i];
    __syncthreads();
    for (int p = tid; p < CC * CC; p += 256) {
        int c = p >> 5, k = p & 31;
        float a = 0.0f;
        for (int e = 0; e < DD / 2; ++e) a += sn[c * (DD / 2) + e] * sn[k * (DD / 2) + e];
        a *= 0.125f;   // 1/sqrt(64)
        P[(long)b * CC * CC + p] = 1.0f / (1.0f + __expf(-a));
    }
}

// xg[b,l,c,d] = sum_k P[b,c,k] * x[b,l,k,d]; output contiguous as (B*L, C, D).
// grid.x = B*L, block = 256.
__global__ void k_xg(const float* __restrict__ x, const float* __restrict__ P,
                     float* __restrict__ XG)
{
    __shared__ float sx[CC * DD];   // 16 KB
    __shared__ float sp[CC * CC];   //  4 KB
    const int bl  = blockIdx.x;
    const int b   = bl / LL;
    const int tid = threadIdx.x;
    for (int i = tid; i < CC * DD; i += 256) sx[i] = x[(long)bl * CC * DD + i];
    for (int i = tid; i < CC * CC; i += 256) sp[i] = P[(long)b * CC * CC + i];
    __syncthreads();
    for (int i = tid; i < CC * DD; i += 256) {
        int c = i >> 7, d = i & 127;
        float a = 0.0f;
#pragma unroll 8
        for (int k = 0; k < CC; ++k) a += sp[c * CC + k] * sx[k * DD + d];
        XG[(long)bl * CC * DD + i] = a;
    }
}

// ---------------------------------------------------------------------------
extern "C" void kernel_launch(void* const* d_in, const int* in_sizes, int n_in,
                              void* d_out, int out_size, void* d_ws, size_t ws_size,
                              hipStream_t stream)
{
    const float* x        = (const float*)d_in[0];
    const float* t_w_in   = (const float*)d_in[1];
    const float* t_conv_w = (const float*)d_in[2];
    const float* t_conv_b = (const float*)d_in[3];
    const float* t_wx     = (const float*)d_in[4];
    const float* t_wdt    = (const float*)d_in[5];
    const float* t_bdt    = (const float*)d_in[6];
    const float* t_alog   = (const float*)d_in[7];
    const float* t_d      = (const float*)d_in[8];
    const float* t_w_out  = (const float*)d_in[9];
    const float* c_w_in   = (const float*)d_in[10];
    const float* c_conv_w = (const float*)d_in[11];
    const float* c_conv_b = (const float*)d_in[12];
    const float* c_wx     = (const float*)d_in[13];
    const float* c_wdt    = (const float*)d_in[14];
    const float* c_bdt    = (const float*)d_in[15];
    const float* c_alog   = (const float*)d_in[16];
    const float* c_d      = (const float*)d_in[17];
    const float* c_w_out  = (const float*)d_in[18];
    const float* g_wn     = (const float*)d_in[19];
    const float* g_bn     = (const float*)d_in[20];

    float* out = (float*)d_out;                 // [T_out | C_out], each NTOK*DD
    float* ws  = (float*)d_ws;

    // Workspace layout (floats); buffers reused across the two mamba passes.
    size_t o = 0;
    float* XZ   = ws + o; o += (size_t)NTOK * XZN;    // 16.8M
    float* U    = ws + o; o += (size_t)NTOK * DI;     //  8.4M
    float* DTb  = ws + o; o += (size_t)NTOK * DI;     //  8.4M
    float* BCb  = ws + o; o += (size_t)NTOK * 32;     //  1.0M
    float* YG   = ws + o; o += (size_t)NTOK * DI;     //  8.4M
    float* XG   = ws + o; o += (size_t)NTOK * DD;     //  4.2M
    float* NODE = ws + o; o += (size_t)BB * CC * (DD / 2);
    float* Pb   = ws + o; o += (size_t)BB * CC * CC;
    // total ~47.2M floats (~181 MB)

    // ---- temporal mamba (sequences over L, token order (b,c,l)) ----
    k_gemm_in <<<NTOK / 16, 256, 0, stream>>>(x, t_w_in, XZ, /*gather*/1);
    k_conv_silu<<<NTOK, DI, 0, stream>>>(XZ, t_conv_w, t_conv_b, U, LL);
    k_xproj_dt<<<NTOK, DI, 0, stream>>>(U, t_wx, t_wdt, t_bdt, DTb, BCb);
    k_scan    <<<NTOK / LL, DI, 0, stream>>>(U, DTb, BCb, XZ, t_alog, t_d, YG, LL);
    k_gemm_out<<<NTOK / 16, 256, 0, stream>>>(YG, t_w_out, out, /*scatter*/1);

    // ---- node attention + channel mix ----
    k_node<<<BB * CC, DD, 0, stream>>>(x, g_wn, g_bn, NODE);
    k_attn<<<BB, 256, 0, stream>>>(NODE, Pb);
    k_xg  <<<BB * LL, 256, 0, stream>>>(x, Pb, XG);

    // ---- channel mamba (sequences over C, contiguous (b*l, c, d)) ----
    k_gemm_in <<<NTOK / 16, 256, 0, stream>>>(XG, c_w_in, XZ, 0);
    k_conv_silu<<<NTOK, DI, 0, stream>>>(XZ, c_conv_w, c_conv_b, U, CC);
    k_xproj_dt<<<NTOK, DI, 0, stream>>>(U, c_wx, c_wdt, c_bdt, DTb, BCb);
    k_scan    <<<NTOK / CC, DI, 0, stream>>>(U, DTb, BCb, XZ, c_alog, c_d, YG, CC);
    k_gemm_out<<<NTOK / 16, 256, 0, stream>>>(YG, c_w_out, out + (size_t)NTOK * DD, 0);
}